// HexPlaneField_4183298146775
// MI455X (gfx1250) — compile-verified
//
#include <hip/hip_runtime.h>

// ---------------------------------------------------------------------------
// HexPlane field sampling for MI455X (gfx1250, wave32).
//  - Kernel 1: repack 18 planes [C=32,H,W] -> [H,W,C=32] into d_ws so each
//    corner gather by a wave is one fully-utilized 128B line.
//  - Kernel 2: one wave32 per point, lane = channel. All index/address math is
//    wave-uniform -> SALU pipe (readfirstlane + scalar adds), co-issuing with
//    the VALU bilinear math; gathers use the saddr+lane form.
// ---------------------------------------------------------------------------

#define NPLANES 18
#define OUTDIM  32

struct PlanePtrs { const float* p[NPLANES]; };

// Compile-time plane geometry (reference: RESO_BASE=[64,64,64,150],
// MULTIRES=[1,2,4], COO_COMBS = (0,1),(0,2),(0,3),(1,2),(1,3),(2,3)).
__host__ __device__ constexpr int  kMult(int l)   { return l == 0 ? 1 : (l == 1 ? 2 : 4); }
__host__ __device__ constexpr int  kW(int pl)     { return 64 * kMult(pl / 6); }       // cc[0] in {0,1,2}
__host__ __device__ constexpr int  kH(int pl) {
    const int ci = pl % 6;
    return (ci == 2 || ci == 4 || ci == 5) ? 150 : 64 * kMult(pl / 6);                 // cc[1]==3 -> 150
}
__host__ __device__ constexpr long kElems(int pl) { return (long)OUTDIM * kW(pl) * kH(pl); }
__host__ __device__ constexpr long kOff(int pl) {
    long o = 0;
    for (int i = 0; i < pl; ++i) o += kElems(i);
    return o;
}

// ---------------------------------------------------------------------------
// Kernel 1: channel-major -> channel-last repack (~59 MB total, L2-resident
// afterwards). global_prefetch_b8 runs ahead of the strided reads.
// ---------------------------------------------------------------------------
__global__ __launch_bounds__(256) void hex_repack(PlanePtrs src, float* __restrict__ dst) {
    const int pl = blockIdx.y;
    int W = 0, H = 0; long off = 0;
#pragma unroll
    for (int i = 0; i < NPLANES; ++i)
        if (i == pl) { W = kW(i); H = kH(i); off = kOff(i); }

    const int   hw    = W * H;
    const int   total = OUTDIM * hw;
    const float* __restrict__ s = src.p[pl];
    float*       __restrict__ d = dst + off;
    const int stride = gridDim.x * blockDim.x;   // multiple of 32 -> channel invariant

    for (int i = blockIdx.x * blockDim.x + threadIdx.x; i < total; i += stride) {
        const int c    = i & (OUTDIM - 1);
        const int sidx = i >> 5;                 // y*W + x
        const int ni   = i + stride;
        if (ni < total)                          // prefetch next iteration's source line
            __builtin_prefetch(&s[(ni & (OUTDIM - 1)) * hw + (ni >> 5)], 0, 0);
        d[i] = s[c * hw + sidx];                 // dst[(y*W+x)*32 + c]
    }
}

// ---------------------------------------------------------------------------
// 1D axis setup (align_corners=True, border padding). The point coordinate is
// wave-uniform, so the integer index is broadcast to an SGPR via readfirstlane
// (exact: all lanes hold the same value); i1 is then derived on the SALU.
// Only the fractional weight stays in a VGPR (no scalar float pipe).
// ---------------------------------------------------------------------------
struct Ax { int i0; int i1; float w; };

template <int D>
__device__ inline Ax axis1d(float v) {
    const float s  = 0.5f * (float)(D - 1);
    float iv = fminf(fmaxf(fmaf(v, s, s), 0.0f), (float)(D - 1));  // fma + clamp
    float f0 = floorf(iv);
    Ax a;
    a.w  = iv - f0;
    a.i0 = __builtin_amdgcn_readfirstlane((int)f0);   // -> SGPR
    a.i1 = (a.i0 + 1 < D) ? a.i0 + 1 : D - 1;         // s_add / s_cselect
    return a;
}

// Bilinear sample of one channel-last plane [H, W, 32]. Row/col offsets and
// the four corner bases are scalar (SALU); each gather is a single coalesced
// 128B line: global_load_b32 v, v_lane, s[corner_base].
template <int W, int H>
__device__ inline float sample_t(const float* __restrict__ g, const Ax& ax, const Ax& ay, int lane) {
    const int r0 = ay.i0 * (W * OUTDIM);     // s_mul
    const int r1 = ay.i1 * (W * OUTDIM);
    const int c0 = ax.i0 * OUTDIM;           // s_lshl
    const int c1 = ax.i1 * OUTDIM;
    const float g00 = (g + (r0 + c0))[lane];
    const float g01 = (g + (r0 + c1))[lane];
    const float g10 = (g + (r1 + c0))[lane];
    const float g11 = (g + (r1 + c1))[lane];
    const float top = fmaf(ax.w, g01 - g00, g00);
    const float bot = fmaf(ax.w, g11 - g10, g10);
    return fmaf(ay.w, bot - top, top);
}

// Fallback sampler for the original [32, H, W] layout (lane-varying base).
template <int W, int H>
__device__ inline float sample_raw(const float* __restrict__ g, const Ax& ax, const Ax& ay, int lane) {
    const float* gc = g + lane * (W * H);
    const float g00 = gc[ay.i0 * W + ax.i0];
    const float g01 = gc[ay.i0 * W + ax.i1];
    const float g10 = gc[ay.i1 * W + ax.i0];
    const float g11 = gc[ay.i1 * W + ax.i1];
    const float top = fmaf(ax.w, g01 - g00, g00);
    const float bot = fmaf(ax.w, g11 - g10, g10);
    return fmaf(ay.w, bot - top, top);
}

// ---------------------------------------------------------------------------
// Kernel 2: wave32 per point, lane = channel.
// ---------------------------------------------------------------------------
template <bool TRANS>
__global__ __launch_bounds__(256) void hex_sample(const float* __restrict__ pts,
                                                  const float* __restrict__ ts,
                                                  PlanePtrs pl,
                                                  const float* __restrict__ aabb,
                                                  float* __restrict__ out, int npts) {
    const int lane = threadIdx.x & 31;
    const int n    = (blockIdx.x * blockDim.x + threadIdx.x) >> 5;   // wave-uniform
    if (n >= npts) return;                                           // whole-wave exit

    const float lo0 = aabb[0], lo1 = aabb[1], lo2 = aabb[2];
    const float hi0 = aabb[3], hi1 = aabb[4], hi2 = aabb[5];
    // p = (pts - aabb[0]) * (2 / (aabb[1] - aabb[0])) - 1 ; p4 = [p, t]
    const float px = (__builtin_nontemporal_load(&pts[3 * (size_t)n + 0]) - lo0) * (2.0f / (hi0 - lo0)) - 1.0f;
    const float py = (__builtin_nontemporal_load(&pts[3 * (size_t)n + 1]) - lo1) * (2.0f / (hi1 - lo1)) - 1.0f;
    const float pz = (__builtin_nontemporal_load(&pts[3 * (size_t)n + 2]) - lo2) * (2.0f / (hi2 - lo2)) - 1.0f;
    const float pt = __builtin_nontemporal_load(&ts[n]);

    // Time axis (H = 150) is identical for all 9 temporal planes: hoisted.
    const Ax at = axis1d<150>(pt);

    float* __restrict__ o = out + (size_t)n * (3 * OUTDIM);

#pragma unroll
    for (int l = 0; l < 3; ++l) {
        float acc;
        if (l == 0) {
            const Ax a0 = axis1d<64>(px), a1 = axis1d<64>(py), a2 = axis1d<64>(pz);
            if (TRANS) {
                acc  = sample_t<64, 64 >(pl.p[0], a0, a1, lane);
                acc *= sample_t<64, 64 >(pl.p[1], a0, a2, lane);
                acc *= sample_t<64, 150>(pl.p[2], a0, at, lane);
                acc *= sample_t<64, 64 >(pl.p[3], a1, a2, lane);
                acc *= sample_t<64, 150>(pl.p[4], a1, at, lane);
                acc *= sample_t<64, 150>(pl.p[5], a2, at, lane);
            } else {
                acc  = sample_raw<64, 64 >(pl.p[0], a0, a1, lane);
                acc *= sample_raw<64, 64 >(pl.p[1], a0, a2, lane);
                acc *= sample_raw<64, 150>(pl.p[2], a0, at, lane);
                acc *= sample_raw<64, 64 >(pl.p[3], a1, a2, lane);
                acc *= sample_raw<64, 150>(pl.p[4], a1, at, lane);
                acc *= sample_raw<64, 150>(pl.p[5], a2, at, lane);
            }
        } else if (l == 1) {
            const Ax a0 = axis1d<128>(px), a1 = axis1d<128>(py), a2 = axis1d<128>(pz);
            if (TRANS) {
                acc  = sample_t<128, 128>(pl.p[6],  a0, a1, lane);
                acc *= sample_t<128, 128>(pl.p[7],  a0, a2, lane);
                acc *= sample_t<128, 150>(pl.p[8],  a0, at, lane);
                acc *= sample_t<128, 128>(pl.p[9],  a1, a2, lane);
                acc *= sample_t<128, 150>(pl.p[10], a1, at, lane);
                acc *= sample_t<128, 150>(pl.p[11], a2, at, lane);
            } else {
                acc  = sample_raw<128, 128>(pl.p[6],  a0, a1, lane);
                acc *= sample_raw<128, 128>(pl.p[7],  a0, a2, lane);
                acc *= sample_raw<128, 150>(pl.p[8],  a0, at, lane);
                acc *= sample_raw<128, 128>(pl.p[9],  a1, a2, lane);
                acc *= sample_raw<128, 150>(pl.p[10], a1, at, lane);
                acc *= sample_raw<128, 150>(pl.p[11], a2, at, lane);
            }
        } else {
            const Ax a0 = axis1d<256>(px), a1 = axis1d<256>(py), a2 = axis1d<256>(pz);
            if (TRANS) {
                acc  = sample_t<256, 256>(pl.p[12], a0, a1, lane);
                acc *= sample_t<256, 256>(pl.p[13], a0, a2, lane);
                acc *= sample_t<256, 150>(pl.p[14], a0, at, lane);
                acc *= sample_t<256, 256>(pl.p[15], a1, a2, lane);
                acc *= sample_t<256, 150>(pl.p[16], a1, at, lane);
                acc *= sample_t<256, 150>(pl.p[17], a2, at, lane);
            } else {
                acc  = sample_raw<256, 256>(pl.p[12], a0, a1, lane);
                acc *= sample_raw<256, 256>(pl.p[13], a0, a2, lane);
                acc *= sample_raw<256, 150>(pl.p[14], a0, at, lane);
                acc *= sample_raw<256, 256>(pl.p[15], a1, a2, lane);
                acc *= sample_raw<256, 150>(pl.p[16], a1, at, lane);
                acc *= sample_raw<256, 150>(pl.p[17], a2, at, lane);
            }
        }
        // 384 MB output stream: nontemporal so the grids stay L2-resident.
        __builtin_nontemporal_store(acc, &o[l * OUTDIM + lane]);
    }
}

// ---------------------------------------------------------------------------
extern "C" void kernel_launch(void* const* d_in, const int* in_sizes, int n_in,
                              void* d_out, int out_size, void* d_ws, size_t ws_size,
                              hipStream_t stream) {
    const float* pts = (const float*)d_in[0];
    const float* ts  = (const float*)d_in[1];
    PlanePtrs raw;
    for (int i = 0; i < NPLANES; ++i) raw.p[i] = (const float*)d_in[2 + i];
    const float* aabb = (const float*)d_in[2 + NPLANES];
    float*       out  = (float*)d_out;

    const int npts    = in_sizes[0] / 3;
    const int threads = 256;                                   // 8 wave32 per block
    const int blocks  = (int)(((long)npts * 32 + threads - 1) / threads);

    constexpr size_t need = (size_t)kOff(NPLANES) * sizeof(float);  // ~58.8 MB
    if (ws_size >= need) {
        hex_repack<<<dim3(2048, NPLANES), 256, 0, stream>>>(raw, (float*)d_ws);
        PlanePtrs tp;
        for (int i = 0; i < NPLANES; ++i) tp.p[i] = (const float*)d_ws + kOff(i);
        hex_sample<true><<<blocks, threads, 0, stream>>>(pts, ts, tp, aabb, out, npts);
    } else {
        hex_sample<false><<<blocks, threads, 0, stream>>>(pts, ts, raw, aabb, out, npts);
    }
}